// PCompanion_79139067396248
// MI455X (gfx1250) — compile-verified
//
#include <hip/hip_runtime.h>
#include <hip/hip_bf16.h>

// ---------------------------------------------------------------------------
// PCompanion pipeline for MI455X (gfx1250, wave32, WMMA).
// All GEMMs use V_WMMA_F32_16X16X4_F32 (exact f32; sims feeds top-k and
// top_vals is an output, so no precision downcast). sims GEMM (52 GFLOP,
// comp_type_table L2-resident at 12.8 MB) is fused with per-row top-8 so the
// 1.6 GB sims matrix is never materialized.
//
// Round-3 change: sims workgroups process 32 rows (was 16). Each B fragment
// now feeds two independent WMMA accumulator chains -> 2x arithmetic
// intensity per load, and L2 re-stream traffic halves (256 WGs x 12.8 MB).
// ---------------------------------------------------------------------------

typedef __attribute__((ext_vector_type(2))) float v2f;
typedef __attribute__((ext_vector_type(8))) float v8f;

#define B_ROWS      8192
#define FEAT        128
#define PEMB        128
#define TEMB        64
#define HID         256
#define NTYPES      50000
#define KTOP        8
#define SIMROWS     32    // rows per sims workgroup
#define SSTR        148   // LDS sims-tile row stride (floats), bank-friendly

__device__ __forceinline__ v8f wmma4(v2f a, v2f b, v8f c) {
  // D = A(16x4,f32) * B(4x16,f32) + C(16x16,f32)
  return __builtin_amdgcn_wmma_f32_16x16x4_f32(
      /*neg_a=*/false, a, /*neg_b=*/false, b,
      /*c_mod=*/(short)0, c, /*reuse_a=*/false, /*reuse_b=*/false);
}

// ---------------------------------------------------------------------------
// Generic wave-per-tile GEMM: C[M,N] = act(A[M,K(lda)] @ W[K,N] + bias[N])
// One 16x16 output tile per wave; 8 waves (256 threads) per workgroup.
// K-loop chunked by 16: 8 loads batched into a clause, then 4 chained WMMAs.
// ---------------------------------------------------------------------------
template <bool RELU>
__global__ __launch_bounds__(256)
void gemm_bias_kernel(const float* __restrict__ A, const float* __restrict__ W,
                      const float* __restrict__ bias, float* __restrict__ C,
                      int N, int K, int lda, int nTiles, int totalTiles) {
  const int wave = threadIdx.x >> 5;
  const int tile = blockIdx.x * 8 + wave;
  if (tile >= totalTiles) return;            // wave-uniform guard
  const int mT = tile / nTiles;
  const int nT = tile - mT * nTiles;
  const int l  = threadIdx.x & 31;
  const int lm = l & 15;                     // M (A) / N (B) within tile
  const int lh = l >> 4;                     // K-half selector
  const int row = mT * 16 + lm;
  const int col = nT * 16 + lm;

  const float* aPtr = A + (size_t)row * lda;
  v8f acc = {};
  for (int k = 0; k < K; k += 16) {          // K always a multiple of 16 here
    v2f a[4], b[4];
#pragma unroll
    for (int j = 0; j < 4; ++j) {
      const int kk = k + 4 * j + 2 * lh;
      a[j].x = aPtr[kk];                  a[j].y = aPtr[kk + 1];
      b[j].x = W[(size_t)kk * N + col];   b[j].y = W[(size_t)(kk + 1) * N + col];
    }
#pragma unroll
    for (int j = 0; j < 4; ++j) acc = wmma4(a[j], b[j], acc);
  }

  const float bv = bias[col];
#pragma unroll
  for (int v = 0; v < 8; ++v) {
    float x = acc[v] + bv;
    if (RELU) x = fmaxf(x, 0.0f);
    C[(size_t)(mT * 16 + v + 8 * lh) * N + col] = x;
  }
}

// ---------------------------------------------------------------------------
// Gather query type embeddings: QT[i,:] = table[qtypes[i],:]
// ---------------------------------------------------------------------------
__global__ __launch_bounds__(256)
void gather_qt_kernel(const int* __restrict__ qtypes,
                      const float* __restrict__ table,
                      float* __restrict__ QT) {
  const int i = blockIdx.x * 256 + threadIdx.x;   // over B_ROWS*TEMB
  const int r = i >> 6;
  const int c = i & 63;
  QT[i] = table[(size_t)qtypes[r] * TEMB + c];
}

// ---------------------------------------------------------------------------
// Register top-8 insertion (descending list, fully unrolled -> stays in VGPRs)
// ---------------------------------------------------------------------------
__device__ __forceinline__ void topk_insert(float (&tv)[KTOP], int (&ti)[KTOP],
                                            float v, int idx) {
  if (v <= tv[KTOP - 1]) return;
  tv[KTOP - 1] = v;
  ti[KTOP - 1] = idx;
#pragma unroll
  for (int s = KTOP - 1; s > 0; --s) {
    if (tv[s] > tv[s - 1]) {
      float fv = tv[s]; tv[s] = tv[s - 1]; tv[s - 1] = fv;
      int   iv = ti[s]; ti[s] = ti[s - 1]; ti[s - 1] = iv;
    } else break;
  }
}

// ---------------------------------------------------------------------------
// Fused sims + top-k: 32 rows/workgroup, 8 waves sweep 128 columns/iteration.
// Two A-fragment sets (rows r0..r0+15, r0+16..r0+31) stay in registers for
// the whole 50000-column sweep; each 16-fragment B load clause feeds two
// independent WMMA accumulator chains.
// ---------------------------------------------------------------------------
__global__ __launch_bounds__(256)
void sims_topk_kernel(const float* __restrict__ CB,   // [B,64] comp_base
                      const float* __restrict__ CTT,  // [NTYPES,64]
                      float* __restrict__ outVals,    // [B,8]
                      float* __restrict__ outIdxF,    // [B,8] (indices as f32)
                      int*   __restrict__ outIdxI) {  // [B,8] (for proj kernel)
  __shared__ float simTile[SIMROWS * SSTR];
  __shared__ float cVal[SIMROWS * 64];
  __shared__ int   cIdx[SIMROWS * 64];

  const int t    = threadIdx.x;
  const int wave = t >> 5;
  const int l    = t & 31;
  const int lm   = l & 15;
  const int lh   = l >> 4;
  const int r0   = blockIdx.x * SIMROWS;

  // A fragments: two 16-row sets, K=64 -> 2 x 16 x v2f per lane (64 VGPRs).
  v2f af0[16], af1[16];
  {
    const float* cb0 = CB + (size_t)(r0 + lm) * TEMB;
    const float* cb1 = CB + (size_t)(r0 + 16 + lm) * TEMB;
#pragma unroll
    for (int i = 0; i < 16; ++i) {
      const int kk = 4 * i + 2 * lh;
      af0[i].x = cb0[kk];  af0[i].y = cb0[kk + 1];
      af1[i].x = cb1[kk];  af1[i].y = cb1[kk + 1];
    }
  }

  // Per-thread top-8 over its (row, column-slot) stream.
  float tv[KTOP]; int ti[KTOP];
#pragma unroll
  for (int k = 0; k < KTOP; ++k) { tv[k] = -3.4e38f; ti[k] = 0; }
  const int selRow  = t >> 3;   // 0..31
  const int selSlot = t & 7;    // 0..7

  for (int j0 = 0; j0 < NTYPES; j0 += 128) {
    const int j = j0 + wave * 16;
    if (j < NTYPES) {                         // wave-uniform
      const int col  = j + lm;
      const int tcol = col < NTYPES ? col : NTYPES - 1;
      const float* bp = CTT + (size_t)tcol * TEMB;
      if (tcol + 128 < NTYPES)                // hint next sweep's tile (L2)
        __builtin_prefetch(bp + 128 * TEMB, 0, 1);
      // Batch all 16 B fragments into distinct regs -> one load clause.
      v2f bf[16];
#pragma unroll
      for (int i = 0; i < 16; ++i) {
        const int kk = 4 * i + 2 * lh;
        bf[i].x = bp[kk];
        bf[i].y = bp[kk + 1];
      }
      // Two independent accumulation chains share each B fragment.
      v8f acc0 = {}, acc1 = {};
#pragma unroll
      for (int i = 0; i < 16; ++i) {
        acc0 = wmma4(af0[i], bf[i], acc0);
        acc1 = wmma4(af1[i], bf[i], acc1);
      }
#pragma unroll
      for (int v = 0; v < 8; ++v) {
        simTile[(v + 8 * lh) * SSTR + wave * 16 + lm]        = acc0[v];
        simTile[(16 + v + 8 * lh) * SSTR + wave * 16 + lm]   = acc1[v];
      }
    }
    __syncthreads();
#pragma unroll
    for (int q = 0; q < 16; ++q) {
      const int c  = selSlot + 8 * q;
      const int gc = j0 + c;
      if (gc < NTYPES)
        topk_insert(tv, ti, simTile[selRow * SSTR + c], gc);
    }
    __syncthreads();
  }

  // Dump per-thread candidates (8 per thread, 64 per row), merge 64 -> 8.
#pragma unroll
  for (int k = 0; k < KTOP; ++k) {
    cVal[selRow * 64 + selSlot * 8 + k] = tv[k];
    cIdx[selRow * 64 + selSlot * 8 + k] = ti[k];
  }
  __syncthreads();

  if (t < SIMROWS) {
    float bv[KTOP]; int bi[KTOP];
#pragma unroll
    for (int k = 0; k < KTOP; ++k) { bv[k] = -3.4e38f; bi[k] = 0; }
    for (int c = 0; c < 64; ++c)
      topk_insert(bv, bi, cVal[t * 64 + c], cIdx[t * 64 + c]);
    const int row = r0 + t;
#pragma unroll
    for (int k = 0; k < KTOP; ++k) {
      outVals[row * KTOP + k] = bv[k];
      outIdxF[row * KTOP + k] = (float)bi[k];
      outIdxI[row * KTOP + k] = bi[k];
    }
  }
}

// ---------------------------------------------------------------------------
// Item projection: out[b,k,:] = concat(QE[b], CTT[top_idx[b,k]]) @ W_ip + b_ip
// Virtual rows R = B*KTOP = 65536; A built on the fly (128-boundary is even,
// so a K-pair never straddles the concat seam). K chunked by 16.
// ---------------------------------------------------------------------------
__global__ __launch_bounds__(256)
void proj_kernel(const float* __restrict__ QE,    // [B,128]
                 const float* __restrict__ CTT,   // [NTYPES,64]
                 const int*   __restrict__ topIdx,// [B*8]
                 const float* __restrict__ Wip,   // [192,128]
                 const float* __restrict__ bip,   // [128]
                 float* __restrict__ outP) {      // [B*8,128]
  const int wave = threadIdx.x >> 5;
  const int tile = blockIdx.x * 8 + wave;         // 32768 tiles total
  const int mT = tile >> 3;                       // R/16 = 4096 row tiles
  const int nT = tile & 7;                        // 128/16 = 8 col tiles
  const int l  = threadIdx.x & 31;
  const int lm = l & 15;
  const int lh = l >> 4;

  const int row  = mT * 16 + lm;                  // virtual row (b*8 + k)
  const int b    = row >> 3;
  const int tIdx = topIdx[row];
  const float* qe = QE + (size_t)b * PEMB;
  const float* te = CTT + (size_t)tIdx * TEMB;
  const int col = nT * 16 + lm;

  v8f acc = {};
#pragma unroll
  for (int kc = 0; kc < 12; ++kc) {               // K = 192 = 12 chunks of 16
    v2f a[4], w[4];
#pragma unroll
    for (int j = 0; j < 4; ++j) {
      const int kk = kc * 16 + 4 * j + 2 * lh;
      if (kk < PEMB) { a[j].x = qe[kk];        a[j].y = qe[kk + 1]; }
      else           { a[j].x = te[kk - PEMB]; a[j].y = te[kk - PEMB + 1]; }
      w[j].x = Wip[(size_t)kk * PEMB + col];
      w[j].y = Wip[(size_t)(kk + 1) * PEMB + col];
    }
#pragma unroll
    for (int j = 0; j < 4; ++j) acc = wmma4(a[j], w[j], acc);
  }

  const float bv = bip[col];
#pragma unroll
  for (int v = 0; v < 8; ++v)
    outP[(size_t)(mT * 16 + v + 8 * lh) * PEMB + col] = acc[v] + bv;
}

// ---------------------------------------------------------------------------
extern "C" void kernel_launch(void* const* d_in, const int* in_sizes, int n_in,
                              void* d_out, int out_size, void* d_ws, size_t ws_size,
                              hipStream_t stream) {
  const float* qf    = (const float*)d_in[0];
  const int*   qty   = (const int*)  d_in[1];
  const float* W1p   = (const float*)d_in[2];
  const float* b1p   = (const float*)d_in[3];
  const float* W2p   = (const float*)d_in[4];
  const float* b2p   = (const float*)d_in[5];
  const float* qtt   = (const float*)d_in[6];   // query_type_table
  const float* ctt   = (const float*)d_in[7];   // comp_type_table
  const float* W1t   = (const float*)d_in[8];
  const float* b1t   = (const float*)d_in[9];
  const float* W2t   = (const float*)d_in[10];
  const float* b2t   = (const float*)d_in[11];
  const float* Wip   = (const float*)d_in[12];
  const float* bip   = (const float*)d_in[13];

  // Workspace carve-up (floats): ~25.4 MB total.
  float* H1  = (float*)d_ws;                       // [B,256]
  float* QE  = H1 + (size_t)B_ROWS * HID;          // [B,128]
  float* QT  = QE + (size_t)B_ROWS * PEMB;         // [B,64]
  float* H2  = QT + (size_t)B_ROWS * TEMB;         // [B,256]
  float* CB  = H2 + (size_t)B_ROWS * HID;          // [B,64]
  int*   IDX = (int*)(CB + (size_t)B_ROWS * TEMB); // [B,8]

  float* outProj = (float*)d_out;                              // [B,8,128]
  float* outIdxF = outProj + (size_t)B_ROWS * KTOP * PEMB;     // [B,8]
  float* outVals = outIdxF + (size_t)B_ROWS * KTOP;            // [B,8]

  const int mT = B_ROWS / 16;  // 512

  // Product2Vec MLP
  {
    int nTiles = HID / 16, total = mT * nTiles;                // 8192
    gemm_bias_kernel<true><<<(total + 7) / 8, 256, 0, stream>>>(
        qf, W1p, b1p, H1, HID, FEAT, FEAT, nTiles, total);
  }
  {
    int nTiles = PEMB / 16, total = mT * nTiles;               // 4096
    gemm_bias_kernel<false><<<(total + 7) / 8, 256, 0, stream>>>(
        H1, W2p, b2p, QE, PEMB, HID, HID, nTiles, total);
  }

  // Type embedding gather + transition MLP
  gather_qt_kernel<<<(B_ROWS * TEMB) / 256, 256, 0, stream>>>(qty, qtt, QT);
  {
    int nTiles = HID / 16, total = mT * nTiles;                // 8192
    gemm_bias_kernel<true><<<(total + 7) / 8, 256, 0, stream>>>(
        QT, W1t, b1t, H2, HID, TEMB, TEMB, nTiles, total);
  }
  {
    int nTiles = TEMB / 16, total = mT * nTiles;               // 2048
    gemm_bias_kernel<false><<<(total + 7) / 8, 256, 0, stream>>>(
        H2, W2t, b2t, CB, TEMB, HID, HID, nTiles, total);
  }

  // Fused sims (B x 50000) + top-8, 32 rows per workgroup
  sims_topk_kernel<<<B_ROWS / SIMROWS, 256, 0, stream>>>(
      CB, ctt, outVals, outIdxF, IDX);

  // Item projection over (B*8) x 192 @ 192 x 128
  {
    int total = (B_ROWS * KTOP / 16) * (PEMB / 16);            // 32768
    proj_kernel<<<total / 8, 256, 0, stream>>>(QE, ctt, IDX, Wip, bip, outProj);
  }
}